// MultiHeadAttentionQuantum_65481071409110
// MI455X (gfx1250) — compile-verified
//
#include <hip/hip_runtime.h>
#include <math.h>

typedef __attribute__((ext_vector_type(2))) float v2f;
typedef __attribute__((ext_vector_type(8))) float v8f;

#define BATCH 8
#define SEQ   256
#define EMB   12
#define HEADS 4
#define HD    3
#define NTOK  (BATCH*SEQ)

// ---------------------------------------------------------------------------
// Single fused kernel: QKV projection -> per-head attention (scores via
// V_WMMA_F32_16X16X4_F32) -> output projection -> analytically-collapsed
// 12-qubit RX/CNOT circuit.
//
// One workgroup per batch (8 blocks x 256 threads = 8 wave32). All
// intermediates (q,k,v,ctx) live in LDS; nothing touches HBM except the
// 96 KB input x, the tiny weights, and the 96 KB output.
//
// Quantum part: RX gates on one wire compose (angle phi_w = attn_out_w +
// rx_theta_w); the pre-CNOT state is a product state of independent bits
// with E[(-1)^b_w] = cos(phi_w); the CNOT chain turns bit w into the prefix
// XOR b_0^..^b_w (w>=1) and bit 0 into b_1^..^b_11, so
//   out[0]    = prod_{j=1..11} cos(phi_j)
//   out[w>=1] = prod_{j=0..w}  cos(phi_j)
// ---------------------------------------------------------------------------
__global__ __launch_bounds__(256)
void mha_quantum_fused_kernel(const float* __restrict__ x,
                              const float* __restrict__ wq,
                              const float* __restrict__ wk,
                              const float* __restrict__ wv,
                              const float* __restrict__ bq,
                              const float* __restrict__ bk,
                              const float* __restrict__ bv,
                              const float* __restrict__ wo,
                              const float* __restrict__ bo,
                              const float* __restrict__ rx_theta,
                              float* __restrict__ out)
{
    const int b = blockIdx.x;
    const int t = threadIdx.x;                    // token within sequence
    const float scale = 0.57735026918962576451f;  // 1/sqrt(hd)

    __shared__ float qs[HEADS][SEQ][4];   // q scaled, padded K=4 (4th = 0)
    __shared__ float ks[HEADS][SEQ][4];
    __shared__ float vs[HEADS][SEQ][4];
    __shared__ float ctxs[SEQ][EMB];      // attention output (pre out-proj)

    // ---------------- Phase 1: full QKV projection for this batch ----------
    const float* xrow = x + (size_t)(b*SEQ + t) * EMB;
    __builtin_prefetch(xrow, 0, 0);               // global_prefetch_b8
    float xr[EMB];
    #pragma unroll
    for (int k = 0; k < EMB; ++k) xr[k] = xrow[k];

    #pragma unroll
    for (int row = 0; row < EMB; ++row) {
        const int hh = row / HD;
        const int jj = row % HD;
        float qa = bq[row], ka = bk[row], va = bv[row];
        #pragma unroll
        for (int k = 0; k < EMB; ++k) {
            qa += xr[k] * wq[row*EMB + k];
            ka += xr[k] * wk[row*EMB + k];
            va += xr[k] * wv[row*EMB + k];
        }
        qs[hh][t][jj] = qa * scale;
        ks[hh][t][jj] = ka;
        vs[hh][t][jj] = va;
    }
    #pragma unroll
    for (int hh = 0; hh < HEADS; ++hh) {
        qs[hh][t][3] = 0.f; ks[hh][t][3] = 0.f; vs[hh][t][3] = 0.f;
    }
    __syncthreads();

    // ---------------- Phase 2: attention, 2 waves per head -----------------
    const int wave = threadIdx.x >> 5;
    const int lane = threadIdx.x & 31;
    const int half = lane >> 4;        // 0: lanes 0-15, 1: lanes 16-31
    const int l15  = lane & 15;
    const int head = wave >> 1;        // 2 wave32 per head
    const int sub  = wave & 1;         // which 128-row half of the sequence

    for (int stripe = 0; stripe < 8; ++stripe) {
        const int row_base = sub*128 + stripe*16;

        // A operand (16x4 f32): lanes 0-15 -> K=0/1 in v0/v1; lanes 16-31 -> K=2/3
        const int am = row_base + l15;
        v2f A;
        A.x = qs[head][am][half ? 2 : 0];
        A.y = qs[head][am][half ? 3 : 1];

        // 16x256 score stripe as 16 WMMA tiles kept in registers
        v8f st[16];
        #pragma unroll
        for (int c = 0; c < 16; ++c) {
            const int s = c*16 + l15;
            // B (4x16 f32): v0 = K0 (lanes 0-15) / K2 (lanes 16-31); v1 = K1/K3
            v2f Bm;
            Bm.x = ks[head][s][half ? 2 : 0];
            Bm.y = ks[head][s][half ? 3 : 1];
            v8f acc = {0.f,0.f,0.f,0.f,0.f,0.f,0.f,0.f};
            st[c] = __builtin_amdgcn_wmma_f32_16x16x4_f32(
                        false, A, false, Bm, (short)0, acc, false, false);
        }

        // Row softmax. C layout: VGPR j = row (row_base + j + 8*half); a
        // row's 256 columns live entirely within this 16-lane half.
        float rsum[8];
        #pragma unroll
        for (int j = 0; j < 8; ++j) {
            float m = st[0][j];
            #pragma unroll
            for (int c = 1; c < 16; ++c) m = fmaxf(m, st[c][j]);
            #pragma unroll
            for (int off = 1; off < 16; off <<= 1)
                m = fmaxf(m, __shfl_xor(m, off, 16));
            float s = 0.f;
            #pragma unroll
            for (int c = 0; c < 16; ++c) {
                float e = expf(st[c][j] - m);
                st[c][j] = e;
                s += e;
            }
            #pragma unroll
            for (int off = 1; off < 16; off <<= 1)
                s += __shfl_xor(s, off, 16);
            rsum[j] = 1.f / s;
        }

        // ctx = P @ V (N=3): per-lane partials + cross-lane reduce
        float a0[8], a1[8], a2[8];
        #pragma unroll
        for (int j = 0; j < 8; ++j) { a0[j] = 0.f; a1[j] = 0.f; a2[j] = 0.f; }
        #pragma unroll
        for (int c = 0; c < 16; ++c) {
            const int s = c*16 + l15;
            const float v0 = vs[head][s][0];
            const float v1 = vs[head][s][1];
            const float v2 = vs[head][s][2];
            #pragma unroll
            for (int j = 0; j < 8; ++j) {
                const float p = st[c][j] * rsum[j];
                a0[j] += p * v0;
                a1[j] += p * v1;
                a2[j] += p * v2;
            }
        }
        #pragma unroll
        for (int j = 0; j < 8; ++j) {
            #pragma unroll
            for (int off = 1; off < 16; off <<= 1) {
                a0[j] += __shfl_xor(a0[j], off, 16);
                a1[j] += __shfl_xor(a1[j], off, 16);
                a2[j] += __shfl_xor(a2[j], off, 16);
            }
        }
        if (l15 == 0) {
            #pragma unroll
            for (int j = 0; j < 8; ++j) {
                const int r = row_base + j + 8*half;
                ctxs[r][head*HD + 0] = a0[j];
                ctxs[r][head*HD + 1] = a1[j];
                ctxs[r][head*HD + 2] = a2[j];
            }
        }
    }
    __syncthreads();

    // ---------------- Phase 3: out-proj + collapsed quantum circuit --------
    float cx[EMB];
    #pragma unroll
    for (int k = 0; k < EMB; ++k) cx[k] = ctxs[t][k];

    float cr[EMB];
    #pragma unroll
    for (int n = 0; n < EMB; ++n) {
        float a = bo[n];
        #pragma unroll
        for (int k = 0; k < EMB; ++k) a += cx[k] * wo[n*EMB + k];
        cr[n] = cosf(a + rx_theta[n]);
    }

    float* o = out + (size_t)(b*SEQ + t) * EMB;
    float p1 = 1.f;
    #pragma unroll
    for (int j = 1; j < EMB; ++j) p1 *= cr[j];
    o[0] = p1;                       // wire 0: XOR of bits 1..11
    float run = cr[0];
    #pragma unroll
    for (int w = 1; w < EMB; ++w) {  // wire w: prefix XOR of bits 0..w
        run *= cr[w];
        o[w] = run;
    }
}

extern "C" void kernel_launch(void* const* d_in, const int* in_sizes, int n_in,
                              void* d_out, int out_size, void* d_ws, size_t ws_size,
                              hipStream_t stream)
{
    const float* x  = (const float*)d_in[0];
    const float* wq = (const float*)d_in[1];
    const float* wk = (const float*)d_in[2];
    const float* wv = (const float*)d_in[3];
    const float* bq = (const float*)d_in[4];
    const float* bk = (const float*)d_in[5];
    const float* bv = (const float*)d_in[6];
    const float* wo = (const float*)d_in[7];
    const float* bo = (const float*)d_in[8];
    const float* th = (const float*)d_in[9];

    (void)d_ws; (void)ws_size; (void)in_sizes; (void)n_in; (void)out_size;

    mha_quantum_fused_kernel<<<BATCH, 256, 0, stream>>>(
        x, wq, wk, wv, bq, bk, bv, wo, bo, th, (float*)d_out);
}